// ClinGRAD_2156073582831
// MI455X (gfx1250) — compile-verified
//
#include <hip/hip_runtime.h>
#include <math.h>

#define HEADS 4
#define CDIM 64
#define HC 256            // HEADS * CDIM
#define NEG_SLOPE 0.2f

typedef __attribute__((ext_vector_type(2)))  float    v2f;
typedef __attribute__((ext_vector_type(8)))  float    v8f;

#ifndef __has_builtin
#define __has_builtin(x) 0
#endif
#if defined(__AMDGCN__) && __has_builtin(__builtin_amdgcn_wmma_f32_16x16x4_f32)
#define USE_WMMA_F32 1
#else
#define USE_WMMA_F32 0
#endif

// ---------------------------------------------------------------------------
// GEMM: C[M,N] = A[M,K] @ B[K,N] (+ bias[N]), all row-major fp32.
// One wave32 per 16x16 tile, K stepped by 4 with V_WMMA_F32_16X16X4_F32.
// f32 WMMA VGPR layout (ISA 7.12.2):
//   A 16x4 : lanes 0-15 hold M=lane, K={0,1} in v0/v1; lanes 16-31 K={2,3}
//   B 4x16 : lanes 0-15 hold N=lane, K={0,1} in v0/v1; lanes 16-31 K={2,3}
//   C 16x16: vgpr i -> row (i + 8*half), col = lane&15
// N must be a multiple of 16 (64 or 256 here). M,K arbitrary.
// OOB A rows read a clamped (valid) row: in WMMA, A-row m only feeds D-row m,
// and OOB D rows are never stored, so no masking is needed -> the main K loop
// has zero predication (no exec churn, loads merge to b64).
// ---------------------------------------------------------------------------
__global__ __launch_bounds__(32)
void k_gemm(const float* __restrict__ A, const float* __restrict__ B,
            const float* __restrict__ bias, float* __restrict__ Cm,
            int M, int K, int N) {
  const int m0   = blockIdx.x * 16;
  const int n0   = blockIdx.y * 16;
  const int lane = threadIdx.x & 31;
  const int half = lane >> 4;
  const int l16  = lane & 15;
  const int col  = n0 + l16;
  const int row  = m0 + l16;

#if USE_WMMA_F32
  const float* __restrict__ Ar = A + (size_t)(row < M ? row : M - 1) * K;
  const float* __restrict__ Bc = B + col;
  v8f acc = {};
  const int K4 = K & ~3;
  int k0 = 0;
  for (; k0 < K4; k0 += 4) {
    const int ka = k0 + 2 * half;
    v2f a, b;
    a.x = Ar[ka];
    a.y = Ar[ka + 1];
    b.x = Bc[(size_t)ka * N];
    b.y = Bc[(size_t)(ka + 1) * N];
    acc = __builtin_amdgcn_wmma_f32_16x16x4_f32(
        false, a, false, b, (short)0, acc, false, false);
  }
  if (k0 < K) {  // K % 4 tail (single iteration, only place with predication)
    const int ka = k0 + 2 * half;
    v2f a, b;
    a.x = (ka     < K) ? Ar[ka]     : 0.0f;
    a.y = (ka + 1 < K) ? Ar[ka + 1] : 0.0f;
    b.x = (ka     < K) ? Bc[(size_t)ka * N]       : 0.0f;
    b.y = (ka + 1 < K) ? Bc[(size_t)(ka + 1) * N] : 0.0f;
    acc = __builtin_amdgcn_wmma_f32_16x16x4_f32(
        false, a, false, b, (short)0, acc, false, false);
  }
  const float bv = bias ? bias[col] : 0.0f;
  if (m0 + 16 <= M) {  // interior tile: unconditional stores
#pragma unroll
    for (int i = 0; i < 8; ++i)
      Cm[(size_t)(m0 + i + 8 * half) * N + col] = acc[i] + bv;
  } else {
#pragma unroll
    for (int i = 0; i < 8; ++i) {
      const int r = m0 + i + 8 * half;
      if (r < M) Cm[(size_t)r * N + col] = acc[i] + bv;
    }
  }
#else
  // Scalar fallback (also what the host-side pass parses): same tiling.
  float acc[8] = {0.f, 0.f, 0.f, 0.f, 0.f, 0.f, 0.f, 0.f};
  for (int k = 0; k < K; ++k) {
    const float bvk = B[(size_t)k * N + col];
#pragma unroll
    for (int i = 0; i < 8; ++i) {
      const int r = m0 + i + 8 * half;
      if (r < M) acc[i] += A[(size_t)r * K + k] * bvk;
    }
  }
#pragma unroll
  for (int i = 0; i < 8; ++i) {
    const int r = m0 + i + 8 * half;
    if (r < M) Cm[(size_t)r * N + col] = acc[i] + (bias ? bias[col] : 0.0f);
  }
#endif
}

// ---------------------------------------------------------------------------
__global__ void k_fill_zero(float* __restrict__ p, size_t n) {
  size_t i = (size_t)blockIdx.x * blockDim.x + threadIdx.x;
  if (i < n) p[i] = 0.0f;
}

__global__ void k_relu(float* __restrict__ p, size_t n) {
  size_t i = (size_t)blockIdx.x * blockDim.x + threadIdx.x;
  if (i < n) p[i] = fmaxf(p[i], 0.0f);
}

// a[n,h] = sum_c h[n, h*64 + c] * att[h*64 + c]
__global__ void k_att_score(const float* __restrict__ hm, const float* __restrict__ att,
                            float* __restrict__ out, int Nn) {
  int i = blockIdx.x * blockDim.x + threadIdx.x;
  if (i >= Nn * HEADS) return;
  int n = i / HEADS, hh = i % HEADS;
  const float* hp = hm + (size_t)n * HC + hh * CDIM;
  const float* ap = att + hh * CDIM;
  float s = 0.0f;
  for (int c = 0; c < CDIM; ++c) s += hp[c] * ap[c];
  out[i] = s;
}

// ce[h] = sum_c We[0, h*64+c] * att_edge[h,c]   (rank-1 edge-attr collapse)
__global__ void k_edge_coef(const float* __restrict__ We, const float* __restrict__ ae,
                            float* __restrict__ ce) {
  int hh = threadIdx.x;
  if (hh < HEADS) {
    float s = 0.0f;
    for (int c = 0; c < CDIM; ++c) s += We[hh * CDIM + c] * ae[hh * CDIM + c];
    ce[hh] = s;
  }
}

// order-preserving float <-> uint key (for atomic segment max)
__device__ __forceinline__ unsigned fkey(float f) {
  unsigned u = __float_as_uint(f);
  return (u >> 31) ? ~u : (u | 0x80000000u);
}
__device__ __forceinline__ float fkey_dec(unsigned k) {
  unsigned u = (k >> 31) ? (k & 0x7FFFFFFFu) : ~k;
  return __uint_as_float(u);
}

// alpha[e,h] = leaky(a_s[src]+a_d[dst] (+ ea*ce)); atomicMax key into mkey[dst,h]
__global__ void k_edge_alpha(const float* __restrict__ a_s, const float* __restrict__ a_d,
                             const int* __restrict__ src, const int* __restrict__ dst,
                             const float* __restrict__ ea, const float* __restrict__ ce,
                             float* __restrict__ alpha, unsigned* __restrict__ mkey, int E) {
  int i = blockIdx.x * blockDim.x + threadIdx.x;
  if (i >= E * HEADS) return;
  int e = i / HEADS, hh = i % HEADS;
  float a = a_s[src[e] * HEADS + hh] + a_d[dst[e] * HEADS + hh];
  if (ea) a += ea[e] * ce[hh];
  a = (a >= 0.0f) ? a : NEG_SLOPE * a;
  alpha[i] = a;
  atomicMax(&mkey[dst[e] * HEADS + hh], fkey(a));
}

// decode keys -> m (sentinel 0 == empty segment -> 0, matching isfinite fixup)
__global__ void k_final_m(unsigned* __restrict__ mk, int n) {
  int i = blockIdx.x * blockDim.x + threadIdx.x;
  if (i < n) {
    unsigned k = mk[i];
    ((float*)mk)[i] = (k == 0u) ? 0.0f : fkey_dec(k);
  }
}

// alpha -> exp(alpha - m[dst]); denom[dst,h] += e
__global__ void k_edge_exp(float* __restrict__ alpha, const float* __restrict__ m,
                           const int* __restrict__ dst, float* __restrict__ denom, int E) {
  int i = blockIdx.x * blockDim.x + threadIdx.x;
  if (i >= E * HEADS) return;
  int e = i / HEADS, hh = i % HEADS;
  float ex = __expf(alpha[i] - m[dst[e] * HEADS + hh]);
  alpha[i] = ex;
  atomicAdd(&denom[dst[e] * HEADS + hh], ex);
}

// acc[dst, :] += hs[src, :] * e[e, head]; one block (64 thr) per edge,
// 128-bit gathers of the source row (bandwidth-bound part of the net).
__global__ __launch_bounds__(64)
void k_edge_aggr(const float* __restrict__ hs, const int* __restrict__ src,
                 const int* __restrict__ dst, const float* __restrict__ ew,
                 float* __restrict__ acc, int E) {
  int e = blockIdx.x;
  if (e >= E) return;
  int t = threadIdx.x;                        // 0..63 ; j = 4t..4t+3
  int s = src[e], d = dst[e];
  float w = ew[e * HEADS + (t >> 4)];         // head = (4t)/64 = t/16
  const float4* hv = (const float4*)(hs + (size_t)s * HC) + t;
  float4 v = *hv;
  float* a = acc + (size_t)d * HC + 4 * t;
  atomicAdd(a + 0, v.x * w);
  atomicAdd(a + 1, v.y * w);
  atomicAdd(a + 2, v.z * w);
  atomicAdd(a + 3, v.w * w);
}

// out[n,j] += acc[n,j] / (denom[n, j/64] + 1e-16) + bias[j]
__global__ void k_rel_final(const float* __restrict__ acc, const float* __restrict__ denom,
                            const float* __restrict__ bias, float* __restrict__ out, int Nn) {
  size_t i = (size_t)blockIdx.x * blockDim.x + threadIdx.x;
  if (i >= (size_t)Nn * HC) return;
  int n = (int)(i / HC), j = (int)(i % HC);
  out[i] += acc[i] / (denom[n * HEADS + (j >> 6)] + 1e-16f) + bias[j];
}

// out[r, c] = x[r,:] @ W[:, c] + b[c]   (c < 3)
__global__ void k_final_lin(const float* __restrict__ x, const float* __restrict__ W,
                            const float* __restrict__ b, float* __restrict__ out, int M) {
  int i = blockIdx.x * blockDim.x + threadIdx.x;
  if (i >= M * 3) return;
  int r = i / 3, c = i % 3;
  float s = b[c];
  for (int k = 0; k < HC; ++k) s += x[(size_t)r * HC + k] * W[k * 3 + c];
  out[i] = s;
}

// ---------------------------------------------------------------------------
// Host-side orchestration
// ---------------------------------------------------------------------------
struct Bip  { const float *Ws, *Wd, *as, *ad, *bias; };
struct Homo { const float *W, *as, *ad, *We, *ae, *bias; };
struct Conv { Bip pg, gp, ps, sp; Homo gg, ss; };
struct GatScratch { float *hs, *hd, *acc, *as, *ad, *mk, *den, *eE, *ce; };

static inline unsigned cdiv(size_t n, unsigned b) { return (unsigned)((n + b - 1) / b); }

static void gemm(hipStream_t st, const float* A, const float* B, const float* bias,
                 float* C, int M, int K, int N) {
  dim3 g(cdiv(M, 16), (unsigned)(N / 16));
  k_gemm<<<g, 32, 0, st>>>(A, B, bias, C, M, K, N);
}

static void run_gat(hipStream_t st,
                    const float* xs, int Ns, const float* xd, int Nd, int Din,
                    const int* src, const int* dst, int E,
                    const float* Ws, const float* Wd,
                    const float* att_s, const float* att_d, const float* bias,
                    const float* ea, const float* We, const float* att_e,
                    float* out, const GatScratch& t, bool homo) {
  gemm(st, xs, Ws, nullptr, t.hs, Ns, Din, HC);
  const float* hd = t.hs;
  if (!homo) { gemm(st, xd, Wd, nullptr, t.hd, Nd, Din, HC); hd = t.hd; }

  k_att_score<<<cdiv((size_t)Ns * HEADS, 256), 256, 0, st>>>(t.hs, att_s, t.as, Ns);
  k_att_score<<<cdiv((size_t)Nd * HEADS, 256), 256, 0, st>>>(hd,   att_d, t.ad, Nd);
  if (ea) k_edge_coef<<<1, HEADS, 0, st>>>(We, att_e, t.ce);

  size_t nh = (size_t)Nd * HEADS;
  k_fill_zero<<<cdiv(nh, 256), 256, 0, st>>>(t.mk,  nh);
  k_fill_zero<<<cdiv(nh, 256), 256, 0, st>>>(t.den, nh);
  size_t na = (size_t)Nd * HC;
  k_fill_zero<<<cdiv(na, 256), 256, 0, st>>>(t.acc, na);

  size_t ne = (size_t)E * HEADS;
  k_edge_alpha<<<cdiv(ne, 256), 256, 0, st>>>(t.as, t.ad, src, dst, ea, t.ce,
                                              t.eE, (unsigned*)t.mk, E);
  k_final_m<<<cdiv(nh, 256), 256, 0, st>>>((unsigned*)t.mk, (int)nh);
  k_edge_exp<<<cdiv(ne, 256), 256, 0, st>>>(t.eE, t.mk, dst, t.den, E);
  k_edge_aggr<<<(unsigned)E, 64, 0, st>>>(t.hs, src, dst, t.eE, t.acc, E);
  k_rel_final<<<cdiv(na, 256), 256, 0, st>>>(t.acc, t.den, bias, out, Nd);
}

struct Edges {
  const int *src_pg, *dst_pg, *src_gg, *dst_gg, *src_ps, *dst_ps, *src_ss, *dst_ss;
  int E_pg, E_gg, E_ps, E_ss;
};

static void run_layer(hipStream_t st, const Conv& cv,
                      const float* pin, const float* gin, const float* sin, int Din,
                      float* pout, float* gout, float* sout,
                      int P, int G, int S, const Edges& ed,
                      const float* ea_gg, const float* ea_ss, const GatScratch& t) {
  size_t np = (size_t)P * HC, ng = (size_t)G * HC, ns = (size_t)S * HC;
  k_fill_zero<<<cdiv(np, 256), 256, 0, st>>>(pout, np);
  k_fill_zero<<<cdiv(ng, 256), 256, 0, st>>>(gout, ng);
  k_fill_zero<<<cdiv(ns, 256), 256, 0, st>>>(sout, ns);
  // og = gat(p->g, pg) + gat(g->g, gg, ea_gg)
  run_gat(st, pin, P, gin, G, Din, ed.src_pg, ed.dst_pg, ed.E_pg,
          cv.pg.Ws, cv.pg.Wd, cv.pg.as, cv.pg.ad, cv.pg.bias,
          nullptr, nullptr, nullptr, gout, t, false);
  run_gat(st, gin, G, gin, G, Din, ed.src_gg, ed.dst_gg, ed.E_gg,
          cv.gg.W, cv.gg.W, cv.gg.as, cv.gg.ad, cv.gg.bias,
          ea_gg, cv.gg.We, cv.gg.ae, gout, t, true);
  // op = gat(g->p, gp, reversed pg edges) + gat(s->p, sp, reversed ps edges)
  run_gat(st, gin, G, pin, P, Din, ed.dst_pg, ed.src_pg, ed.E_pg,
          cv.gp.Ws, cv.gp.Wd, cv.gp.as, cv.gp.ad, cv.gp.bias,
          nullptr, nullptr, nullptr, pout, t, false);
  run_gat(st, sin, S, pin, P, Din, ed.dst_ps, ed.src_ps, ed.E_ps,
          cv.sp.Ws, cv.sp.Wd, cv.sp.as, cv.sp.ad, cv.sp.bias,
          nullptr, nullptr, nullptr, pout, t, false);
  // os = gat(p->s, ps) + gat(s->s, ss, ea_ss)
  run_gat(st, pin, P, sin, S, Din, ed.src_ps, ed.dst_ps, ed.E_ps,
          cv.ps.Ws, cv.ps.Wd, cv.ps.as, cv.ps.ad, cv.ps.bias,
          nullptr, nullptr, nullptr, sout, t, false);
  run_gat(st, sin, S, sin, S, Din, ed.src_ss, ed.dst_ss, ed.E_ss,
          cv.ss.W, cv.ss.W, cv.ss.as, cv.ss.ad, cv.ss.bias,
          ea_ss, cv.ss.We, cv.ss.ae, sout, t, true);
  k_relu<<<cdiv(np, 256), 256, 0, st>>>(pout, np);
  k_relu<<<cdiv(ng, 256), 256, 0, st>>>(gout, ng);
  k_relu<<<cdiv(ns, 256), 256, 0, st>>>(sout, ns);
}

extern "C" void kernel_launch(void* const* d_in, const int* in_sizes, int n_in,
                              void* d_out, int out_size, void* d_ws, size_t ws_size,
                              hipStream_t stream) {
  (void)n_in; (void)out_size; (void)ws_size;
  // Assumed input order: setup_inputs() dict insertion order, with the nested
  // `params` dict flattened depth-first in insertion order (one pointer per leaf).
  int ii = 0;
  auto nf = [&]() { return (const float*)d_in[ii++]; };

  const float* x_p = nf();                 // [P,4]
  const float* x_g = nf();                 // [G,1]
  const float* x_s = nf();                 // [S,107]
  const float* ea_gg = nf();               // [E_gg,1]
  const float* ea_ss = nf();               // [E_ss,1]
  const int P = in_sizes[0] / 4;
  const int G = in_sizes[1];
  const int S = in_sizes[2] / 107;
  const int E_gg = in_sizes[3];
  const int E_ss = in_sizes[4];

  const float* pl_W = nf(); const float* pl_b = nf();   // patient_lin
  const float* gl_W = nf(); const float* gl_b = nf();   // gene_lin
  const float* sl_W = nf(); const float* sl_b = nf();   // structure_lin

  auto rb = [&]() { Bip b; b.Ws = nf(); b.Wd = nf(); b.as = nf(); b.ad = nf(); b.bias = nf(); return b; };
  auto rh = [&]() { Homo h; h.W = nf(); h.as = nf(); h.ad = nf(); h.We = nf(); h.ae = nf(); h.bias = nf(); return h; };
  auto rc = [&]() { Conv c; c.pg = rb(); c.gp = rb(); c.gg = rh(); c.ps = rb(); c.sp = rb(); c.ss = rh(); return c; };
  Conv c1 = rc();
  Conv c2 = rc();
  const float* lin_W = nf(); const float* lin_b = nf();

  Edges ed;
  int i_pg = ii; ed.src_pg = (const int*)d_in[ii++]; ed.dst_pg = (const int*)d_in[ii++];
  ed.src_gg = (const int*)d_in[ii++]; ed.dst_gg = (const int*)d_in[ii++];
  int i_ps = ii; ed.src_ps = (const int*)d_in[ii++]; ed.dst_ps = (const int*)d_in[ii++];
  ed.src_ss = (const int*)d_in[ii++]; ed.dst_ss = (const int*)d_in[ii++];
  ed.E_pg = in_sizes[i_pg]; ed.E_gg = E_gg; ed.E_ps = in_sizes[i_ps]; ed.E_ss = E_ss;
  int Emax = ed.E_pg > ed.E_gg ? ed.E_pg : ed.E_gg;
  if (ed.E_ps > Emax) Emax = ed.E_ps;
  if (ed.E_ss > Emax) Emax = ed.E_ss;
  int Nmax = G > S ? (G > P ? G : P) : (S > P ? S : P);

  // Workspace layout (floats)
  float* w = (float*)d_ws;
  size_t off = 0;
  auto alloc = [&](size_t n) { float* q = w + off; off += n; return q; };
  float* p0 = alloc((size_t)P * 64);
  float* g0 = alloc((size_t)G * 64);
  float* s0 = alloc((size_t)S * 64);
  float* pA = alloc((size_t)P * HC);
  float* gA = alloc((size_t)G * HC);
  float* sA = alloc((size_t)S * HC);
  float* pB = alloc((size_t)P * HC);
  float* gB = alloc((size_t)G * HC);
  float* sB = alloc((size_t)S * HC);
  GatScratch t;
  t.hs  = alloc((size_t)Nmax * HC);
  t.hd  = alloc((size_t)Nmax * HC);
  t.acc = alloc((size_t)Nmax * HC);
  t.as  = alloc((size_t)Nmax * HEADS);
  t.ad  = alloc((size_t)Nmax * HEADS);
  t.mk  = alloc((size_t)Nmax * HEADS);
  t.den = alloc((size_t)Nmax * HEADS);
  t.eE  = alloc((size_t)Emax * HEADS);
  t.ce  = alloc(HEADS);

  // Input projections (fp32 WMMA GEMMs, N=64)
  gemm(stream, x_p, pl_W, pl_b, p0, P, 4, 64);
  gemm(stream, x_g, gl_W, gl_b, g0, G, 1, 64);
  gemm(stream, x_s, sl_W, sl_b, s0, S, 107, 64);

  // Two hetero-GAT layers (ping-pong A/B), then final linear on patients.
  run_layer(stream, c1, p0, g0, s0, 64, pA, gA, sA, P, G, S, ed, ea_gg, ea_ss, t);
  run_layer(stream, c2, pA, gA, sA, HC, pB, gB, sB, P, G, S, ed, ea_gg, ea_ss, t);
  k_final_lin<<<cdiv((size_t)P * 3, 256), 256, 0, stream>>>(pB, lin_W, lin_b,
                                                            (float*)d_out, P);
}